// Mamba3Layer_39144331935958
// MI455X (gfx1250) — compile-verified
//
#include <hip/hip_runtime.h>
#include <math.h>

typedef __attribute__((ext_vector_type(2))) float v2f;
typedef __attribute__((ext_vector_type(8))) float v8f;

#define D_MODEL 1024
#define D_STATE 32
#define D_INNER 2048
#define BSZ 2
#define SEQ 512

#define BM 128
#define BN 64
#define BK 16
#define LDSS 20   // padded LDS row stride: 80B rows -> B128 async stores stay
                  // 16B aligned, v2f fragment reads stay 8B aligned.

// ---------------------------------------------------------------------------
// CDNA5 async global->LDS copy of one 16-byte granule per lane.
// VDST = 32-bit LDS byte address (low 32 bits of the generic shared pointer),
// VADDR = 64-bit global address. Tracked with ASYNCcnt.
// ---------------------------------------------------------------------------
__device__ __forceinline__ void async_copy_b128(const float* g, float* l) {
    asm volatile("global_load_async_to_lds_b128 %0, %1, off"
                 :: "v"((unsigned)(size_t)l), "v"(g)
                 : "memory");
}
__device__ __forceinline__ void wait_async_le(int) {}  // (documentation only)

// ---------------------------------------------------------------------------
// C[M,N] = X[M,K] @ W[N,K]^T   (fp32, V_WMMA_F32_16X16X4_F32)
// block = 256 threads = 8 waves (4 x 2), wave tile = 32x32 (2x2 WMMA tiles).
// Double-buffered LDS fed by GLOBAL_LOAD_ASYNC_TO_LDS_B128.
// ---------------------------------------------------------------------------
__global__ __launch_bounds__(256) void gemm_xwT(const float* __restrict__ X,
                                                const float* __restrict__ W,
                                                float* __restrict__ Cout,
                                                int M, int N, int K) {
    __shared__ float Xs[2][BM * LDSS];
    __shared__ float Ws[2][BN * LDSS];

    const int tid  = threadIdx.x;
    const int wave = tid >> 5;
    const int lane = tid & 31;
    const int wm   = wave >> 1;   // 0..3  (M direction)
    const int wn   = wave & 1;    // 0..1  (N direction)
    const int m0   = blockIdx.y * BM;
    const int n0   = blockIdx.x * BN;
    const int r    = lane & 15;   // fragment row
    const int g    = lane >> 4;   // half-wave group

    v8f acc[2][2];
    #pragma unroll
    for (int i = 0; i < 2; ++i)
        #pragma unroll
        for (int j = 0; j < 2; ++j)
            acc[i][j] = v8f{};

    const int lrow = tid >> 2;          // 0..63
    const int lcol = (tid & 3) << 2;    // 0,4,8,12 (floats -> 16B granules)
    // Clamp (instead of predicating) the W row so every wave always issues
    // exactly 3 async copies per stage -> ASYNCcnt bookkeeping stays fixed.
    // Duplicated rows only feed output columns discarded by the store guard.
    const int gn_c = (n0 + lrow < N) ? (n0 + lrow) : (N - 1);

    // stage tile at k0 into buffer `buf` (3 async B128 copies per thread)
    auto stage = [&](int buf, int k0) {
        #pragma unroll
        for (int h = 0; h < 2; ++h) {
            const int row = lrow + h * 64;
            async_copy_b128(X + (size_t)(m0 + row) * K + k0 + lcol,
                            &Xs[buf][row * LDSS + lcol]);
        }
        async_copy_b128(W + (size_t)gn_c * K + k0 + lcol,
                        &Ws[buf][lrow * LDSS + lcol]);
    };

    int buf = 0;
    stage(0, 0);

    for (int k0 = 0; k0 < K; k0 += BK) {
        if (k0 + BK < K) {
            stage(buf ^ 1, k0 + BK);                       // prefetch next tile
            asm volatile("s_wait_asynccnt 0x3" ::: "memory");  // current tile done
        } else {
            asm volatile("s_wait_asynccnt 0x0" ::: "memory");
        }
        __syncthreads();

        #pragma unroll
        for (int kk = 0; kk < BK; kk += 4) {
            v2f a[2], bf[2];
            #pragma unroll
            for (int i = 0; i < 2; ++i)
                a[i] = *(const v2f*)&Xs[buf][(wm * 32 + i * 16 + r) * LDSS + kk + 2 * g];
            #pragma unroll
            for (int j = 0; j < 2; ++j)
                bf[j] = *(const v2f*)&Ws[buf][(wn * 32 + j * 16 + r) * LDSS + kk + 2 * g];
            #pragma unroll
            for (int i = 0; i < 2; ++i)
                #pragma unroll
                for (int j = 0; j < 2; ++j)
                    acc[i][j] = __builtin_amdgcn_wmma_f32_16x16x4_f32(
                        false, a[i], false, bf[j], (short)0, acc[i][j], false, false);
        }
        __syncthreads();
        buf ^= 1;
    }

    // --- store: D layout lane (n + 16g), VGPR c -> row (c + 8g), col n ---
    #pragma unroll
    for (int i = 0; i < 2; ++i)
        #pragma unroll
        for (int j = 0; j < 2; ++j) {
            const int n = n0 + wn * 32 + j * 16 + r;
            if (n < N) {
                #pragma unroll
                for (int c = 0; c < 8; ++c) {
                    const int m = m0 + wm * 32 + i * 16 + c + 8 * g;
                    Cout[(size_t)m * N + n] = acc[i][j][c];
                }
            }
        }
}

// ---------------------------------------------------------------------------
// Sequential scan: one wave per (b,d) channel, lane s owns state[s].
// RoPE mixing via shfl_xor(16); y = state . C via xor-shuffle reduction.
// ---------------------------------------------------------------------------
__global__ __launch_bounds__(256) void scan_kernel(
    const float* __restrict__ xz, const float* __restrict__ dt_raw,
    const float* __restrict__ Bm, const float* __restrict__ Cm,
    const float* __restrict__ A_log, const float* __restrict__ b_dt,
    const float* __restrict__ dt_bias, const float* __restrict__ rope_freq,
    float* __restrict__ y_ws, float* __restrict__ final_state) {
    const int lane = threadIdx.x & 31;
    const int w    = blockIdx.x * 8 + (threadIdx.x >> 5);
    const int b    = w >> 11;          // / D_INNER
    const int d    = w & (D_INNER - 1);

    const float Aval = -__expf(A_log[d * D_STATE + lane]);
    const float bias = b_dt[d] + dt_bias[d];
    const float freq = rope_freq[lane & 15];

    float state = 0.0f;

    for (int l = 0; l < SEQ; ++l) {
        const size_t row = (size_t)(b * SEQ + l);
        float dtr = dt_raw[row * D_INNER + d] + bias;
        float dt  = (dtr > 20.0f) ? dtr : log1pf(__expf(dtr));
        const float x_t = xz[row * (2 * D_INNER) + d];
        const float zv  = xz[row * (2 * D_INNER) + D_INNER + d];
        const float Bv  = Bm[row * D_STATE + lane];
        const float Cv  = Cm[row * D_STATE + lane];

        const float dtA  = dt * Aval;
        const float den  = 1.0f - 0.5f * dtA;
        const float Abar = (1.0f + 0.5f * dtA) / den;
        const float Bbar = dt / den;
        state = Abar * state + Bbar * Bv * x_t;

        // RoPE rotation: lanes [0,16) = real, [16,32) = imag
        const float ang = (float)l * freq;
        float sa, ca;
        __sincosf(ang, &sa, &ca);
        const float partner = __shfl_xor(state, 16, 32);
        state = (lane < 16) ? (state * ca - partner * sa)
                            : (partner * sa + state * ca);

        // y = dot(state, C) over the 32 lanes
        float y = state * Cv;
        #pragma unroll
        for (int off = 16; off; off >>= 1) y += __shfl_xor(y, off, 32);

        if (lane == 0) {
            const float sig = 1.0f / (1.0f + __expf(-zv));
            y_ws[row * D_INNER + d] = y * (zv * sig);   // gate with silu(z)
        }
    }
    final_state[(size_t)(b * D_INNER + d) * D_STATE + lane] = state;
}

// ---------------------------------------------------------------------------
// In-place LayerNorm over D_INNER per token row. 256 threads, 8 elems/thread.
// ---------------------------------------------------------------------------
__device__ __forceinline__ float block_sum(float s, float* sred, int tid) {
    #pragma unroll
    for (int off = 16; off; off >>= 1) s += __shfl_xor(s, off, 32);
    if ((tid & 31) == 0) sred[tid >> 5] = s;
    __syncthreads();
    if (tid == 0) {
        float t = 0.0f;
        #pragma unroll
        for (int i = 0; i < 8; ++i) t += sred[i];
        sred[0] = t;
    }
    __syncthreads();
    const float t = sred[0];
    __syncthreads();
    return t;
}

__global__ __launch_bounds__(256) void layernorm_kernel(float* __restrict__ y,
                                                        const float* __restrict__ ln_w,
                                                        const float* __restrict__ ln_b) {
    __shared__ float sred[8];
    float* p = y + (size_t)blockIdx.x * D_INNER;
    const int tid = threadIdx.x;

    float v[8];
    float s = 0.0f;
    #pragma unroll
    for (int i = 0; i < 8; ++i) { v[i] = p[tid + i * 256]; s += v[i]; }
    const float mu = block_sum(s, sred, tid) * (1.0f / D_INNER);

    float q = 0.0f;
    #pragma unroll
    for (int i = 0; i < 8; ++i) { const float dv = v[i] - mu; q += dv * dv; }
    const float var = block_sum(q, sred, tid) * (1.0f / D_INNER);
    const float rs  = rsqrtf(var + 1e-5f);

    #pragma unroll
    for (int i = 0; i < 8; ++i) {
        const int idx = tid + i * 256;
        p[idx] = (v[i] - mu) * rs * ln_w[idx] + ln_b[idx];
    }
}

// ---------------------------------------------------------------------------
extern "C" void kernel_launch(void* const* d_in, const int* in_sizes, int n_in,
                              void* d_out, int out_size, void* d_ws, size_t ws_size,
                              hipStream_t stream) {
    (void)in_sizes; (void)n_in; (void)out_size; (void)ws_size;
    const float* x         = (const float*)d_in[0];
    const float* W_in      = (const float*)d_in[1];
    const float* A_log     = (const float*)d_in[2];
    const float* W_B       = (const float*)d_in[3];
    const float* W_C       = (const float*)d_in[4];
    const float* W_dt      = (const float*)d_in[5];
    const float* b_dt      = (const float*)d_in[6];
    const float* dt_bias   = (const float*)d_in[7];
    const float* rope_freq = (const float*)d_in[8];
    const float* ln_w      = (const float*)d_in[9];
    const float* ln_b      = (const float*)d_in[10];
    const float* W_out     = (const float*)d_in[11];

    float* out    = (float*)d_out;
    float* fstate = out + (size_t)BSZ * SEQ * D_MODEL;   // tuple part 2

    // workspace layout (floats)
    float* ws     = (float*)d_ws;
    float* xz     = ws;                                         // 2*512*4096
    float* dt_raw = xz + (size_t)BSZ * SEQ * 2 * D_INNER;       // 2*512*2048
    float* Bmp    = dt_raw + (size_t)BSZ * SEQ * D_INNER;       // 2*512*32
    float* Cmp    = Bmp + (size_t)BSZ * SEQ * D_STATE;          // 2*512*32
    float* y_ws   = Cmp + (size_t)BSZ * SEQ * D_STATE;          // 2*512*2048

    const int M = BSZ * SEQ;   // 1024 token rows
    dim3 blk(256);

    // input projections (all share A = x)
    gemm_xwT<<<dim3((2 * D_INNER) / BN, M / BM), blk, 0, stream>>>(x, W_in, xz, M, 2 * D_INNER, D_MODEL);
    gemm_xwT<<<dim3(D_INNER / BN, M / BM), blk, 0, stream>>>(x, W_dt, dt_raw, M, D_INNER, D_MODEL);
    gemm_xwT<<<dim3(1, M / BM), blk, 0, stream>>>(x, W_B, Bmp, M, D_STATE, D_MODEL);
    gemm_xwT<<<dim3(1, M / BM), blk, 0, stream>>>(x, W_C, Cmp, M, D_STATE, D_MODEL);

    // sequential state-space scan: 4096 waves (one per (b,d))
    scan_kernel<<<dim3(BSZ * D_INNER / 8), blk, 0, stream>>>(
        xz, dt_raw, Bmp, Cmp, A_log, b_dt, dt_bias, rope_freq, y_ws, fstate);

    // gated layernorm (in place) + output projection
    layernorm_kernel<<<dim3(M), blk, 0, stream>>>(y_ws, ln_w, ln_b);
    gemm_xwT<<<dim3(D_MODEL / BN, M / BM), blk, 0, stream>>>(y_ws, W_out, out, M, D_MODEL, D_INNER);
}